// Geometric_loss_88862873354310
// MI455X (gfx1250) — compile-verified
//
#include <hip/hip_runtime.h>
#include <math.h>

typedef float v2f __attribute__((ext_vector_type(2)));
typedef float v8f __attribute__((ext_vector_type(8)));

#define BIGC 1000000000.0f
#define BB 2
#define VV 3
#define WW 64
#define NN 4096

// workspace layout (float offsets)
#define WS_CNT   0      // 6 mask counts
#define WS_KINV  16     // 2 * 9
#define WS_T     48     // 8 * 16 : t2r at 48 + j*16, r2t at 112 + j*16  (j = b*2 + (i-1))
#define WS_PART  256    // 16 tasks * 256 rowtiles partial sums
#define WS_PTS   4352   // 14 point sets * 4096 * 3 floats

__device__ inline void rigid_inv4(const float* P, float* Q) {
    // P is rigid [R t; 0 0 0 1], row-major. Q = P^-1 = [R^T, -R^T t; 0 0 0 1]
    for (int r = 0; r < 3; ++r)
        for (int c = 0; c < 3; ++c)
            Q[r * 4 + c] = P[c * 4 + r];
    for (int r = 0; r < 3; ++r)
        Q[r * 4 + 3] = -(P[0 * 4 + r] * P[3] + P[1 * 4 + r] * P[7] + P[2 * 4 + r] * P[11]);
    Q[12] = 0.f; Q[13] = 0.f; Q[14] = 0.f; Q[15] = 1.f;
}

__device__ inline void mat4mul(const float* A, const float* B, float* C) {
    for (int r = 0; r < 4; ++r)
        for (int c = 0; c < 4; ++c) {
            float s = 0.f;
            for (int k = 0; k < 4; ++k) s += A[r * 4 + k] * B[k * 4 + c];
            C[r * 4 + c] = s;
        }
}

// ---------------- kernel 0: transforms, K^-1, zero counters ----------------
__global__ void prep_kernel(const float* __restrict__ poses,
                            const float* __restrict__ intr,
                            float* __restrict__ ws) {
    int tid = threadIdx.x;
    if (tid < 6) ws[WS_CNT + tid] = 0.f;
    if (tid >= 8 && tid < 10) {
        int b = tid - 8;
        const float* K = intr + b * 9;
        float a = K[0], bb = K[1], c = K[2], d = K[3], e = K[4], f = K[5],
              g = K[6], h = K[7], ii = K[8];
        float A = e * ii - f * h, Bm = -(d * ii - f * g), Cm = d * h - e * g;
        float det = a * A + bb * Bm + c * Cm;
        float inv = 1.f / det;
        float* Ki = ws + WS_KINV + b * 9;
        Ki[0] = A * inv;                Ki[1] = (c * h - bb * ii) * inv;  Ki[2] = (bb * f - c * e) * inv;
        Ki[3] = Bm * inv;               Ki[4] = (a * ii - c * g) * inv;   Ki[5] = (c * d - a * f) * inv;
        Ki[6] = Cm * inv;               Ki[7] = (bb * h - a * g) * inv;   Ki[8] = (a * e - bb * d) * inv;
    }
    if (tid >= 12 && tid < 16) {
        int j = tid - 12;
        int b = j >> 1;
        int i = 1 + (j & 1);
        const float* P0 = poses + (b * VV + 0) * 16;
        const float* Pi = poses + (b * VV + i) * 16;
        float P0inv[16], Piinv[16];
        rigid_inv4(P0, P0inv);
        rigid_inv4(Pi, Piinv);
        mat4mul(P0inv, Pi, ws + WS_T + j * 16);        // T_t2r = inv(pose_ref) * pose_i
        mat4mul(Piinv, P0, ws + WS_T + 64 + j * 16);   // T_r2t = inv(pose_i)  * pose_ref
    }
}

// ---------------- kernel 1: mask counts (exact integer float adds) ----------------
__global__ void mask_count_kernel(const float* __restrict__ masks, float* __restrict__ ws) {
    int bv = blockIdx.x;
    const float* m = masks + (size_t)bv * NN;
    float s = 0.f;
    for (int k = threadIdx.x; k < NN; k += 256) s += m[k];
    atomicAdd(&ws[WS_CNT + bv], s);   // small integers: exact, order-independent
}

// ---------------- kernel 2: generate all 14 point sets ----------------
// sets 0..5  : cam(b,v)            (b = s/3, v = s%3)
// sets 6..9  : T_t2r(b,i) * cam(b,i)
// sets 10..13: T_r2t(b,i) * cam(b,0)
__global__ void points_kernel(const float* __restrict__ depth, float* __restrict__ ws) {
    int gid = blockIdx.x * 256 + threadIdx.x;
    int s = gid >> 12;
    int n = gid & (NN - 1);
    if (s >= 14) return;
    int b, v;
    const float* T = nullptr;
    if (s < 6) { b = s / 3; v = s % 3; }
    else if (s < 10) { int j = s - 6; b = j >> 1; v = 1 + (j & 1); T = ws + WS_T + j * 16; }
    else { int j = s - 10; b = j >> 1; v = 0; T = ws + WS_T + 64 + j * 16; }

    float d = depth[((size_t)(b * VV + v)) * NN + n];
    float px = (float)(n & (WW - 1));
    float py = (float)(n >> 6);
    const float* Ki = ws + WS_KINV + b * 9;
    float cx = (Ki[0] * px + Ki[1] * py + Ki[2]) * d;
    float cy = (Ki[3] * px + Ki[4] * py + Ki[5]) * d;
    float cz = (Ki[6] * px + Ki[7] * py + Ki[8]) * d;
    float ox = cx, oy = cy, oz = cz;
    if (T) {
        ox = T[0] * cx + T[1] * cy + T[2]  * cz + T[3];
        oy = T[4] * cx + T[5] * cy + T[6]  * cz + T[7];
        oz = T[8] * cx + T[9] * cy + T[10] * cz + T[11];
    }
    float* p = ws + WS_PTS + ((size_t)s * NN + n) * 3;
    p[0] = ox; p[1] = oy; p[2] = oz;
}

// ---------------- kernel 3: WMMA chamfer (16 directional tasks) ----------------
// task decode: b = t>>3, i = 1+((t>>2)&1), fb = (t>>1)&1 (0=fwd,1=bwd), dir = t&1 (swap rows/cols)
__global__ void chamfer_kernel(const float* __restrict__ masks, float* __restrict__ ws) {
    int task = blockIdx.x >> 5;
    int blk  = blockIdx.x & 31;
    int wave = threadIdx.x >> 5;
    int lane = threadIdx.x & 31;
    int rowTile = blk * 8 + wave;
    int rowBase = rowTile * 16;

    int b   = task >> 3;
    int i   = 1 + ((task >> 2) & 1);
    int fb  = (task >> 1) & 1;
    int dir = task & 1;

    int aset, bset, maskA, maskB;
    if (fb == 0) { aset = b * 3;     bset = 6  + b * 2 + (i - 1); maskA = b * 3;     maskB = b * 3 + i; }
    else         { aset = b * 3 + i; bset = 10 + b * 2 + (i - 1); maskA = b * 3 + i; maskB = b * 3; }
    int rset = dir ? bset : aset;
    int cset = dir ? aset : bset;
    int rmsk = dir ? maskB : maskA;
    int cmsk = dir ? maskA : maskB;

    const float* rp = ws + WS_PTS + (size_t)rset * NN * 3;
    const float* cp = ws + WS_PTS + (size_t)cset * NN * 3;
    const float* rm = masks + (size_t)rmsk * NN;
    const float* cm = masks + (size_t)cmsk * NN;

    int l16 = lane & 15;
    int hi  = lane >> 4;

    // A fragment: 16x4 f32, lanes 0-15: row M=lane, K={0,1}; lanes 16-31: K={2,3}
    v2f afrag;
    {
        int m = rowBase + l16;
        afrag.x = hi ? rp[m * 3 + 2] : rp[m * 3 + 0];
        afrag.y = hi ? 0.f           : rp[m * 3 + 1];
    }
    // |a_row|^2 for rows this half's C registers correspond to (M = r + 8*hi)
    float ra2[8];
    for (int r = 0; r < 8; ++r) {
        int row = rowBase + r + hi * 8;
        float x = rp[row * 3], y = rp[row * 3 + 1], z = rp[row * 3 + 2];
        ra2[r] = x * x + y * y + z * z;
    }
    float rmin[8];
    for (int r = 0; r < 8; ++r) rmin[r] = 3.4e38f;

    for (int ct = 0; ct < 256; ++ct) {
        int col = ct * 16 + l16;
        float bx = cp[col * 3], by = cp[col * 3 + 1], bz = cp[col * 3 + 2];
        v2f bfrag;   // B 4x16: lane holds column N=lane%16, K pair split by lane half
        bfrag.x = hi ? bz : bx;
        bfrag.y = hi ? 0.f : by;
        float cterm = bx * bx + by * by + bz * bz + (1.f - cm[col]) * BIGC;
        v8f acc = {0.f, 0.f, 0.f, 0.f, 0.f, 0.f, 0.f, 0.f};
        acc = __builtin_amdgcn_wmma_f32_16x16x4_f32(
            /*neg_a=*/false, afrag, /*neg_b=*/false, bfrag,
            /*c_mod=*/(short)0, acc, /*reuse_a=*/false, /*reuse_b=*/false);
        for (int r = 0; r < 8; ++r) {
            float cand = fmaf(-2.f, acc[r], ra2[r] + cterm);
            rmin[r] = fminf(rmin[r], cand);
        }
    }

    // min across the 16 lanes of each half (covers all column residues)
    for (int off = 8; off >= 1; off >>= 1)
        for (int r = 0; r < 8; ++r)
            rmin[r] = fminf(rmin[r], __shfl_xor(rmin[r], off, 32));

    // masked row sum: low half covers rows rowBase+0..7, high half rows +8..15
    float sum = 0.f;
    for (int r = 0; r < 8; ++r) {
        int row = rowBase + r + hi * 8;
        sum += rmin[r] * rm[row];
    }
    float total = sum + __shfl_xor(sum, 16, 32);
    if (lane == 0)
        ws[WS_PART + task * 256 + rowTile] = total;   // unique slot -> deterministic
}

// ---------------- kernel 4: finalize scalar ----------------
__global__ void finalize_kernel(const float* __restrict__ ws, float* __restrict__ out) {
    __shared__ float ts[16];
    int t = threadIdx.x;
    if (t < 16) {
        int b   = t >> 3;
        int i   = 1 + ((t >> 2) & 1);
        int fb  = (t >> 1) & 1;
        int dir = t & 1;
        int maskA = fb ? (b * 3 + i) : (b * 3);
        int maskB = fb ? (b * 3)     : (b * 3 + i);
        int rmsk  = dir ? maskB : maskA;
        float s = 0.f;
        const float* part = ws + WS_PART + t * 256;
        for (int k = 0; k < 256; ++k) s += part[k];
        ts[t] = s / fmaxf(ws[WS_CNT + rmsk], 1.f);
    }
    __syncthreads();
    if (t == 0) {
        float tot = 0.f;
        for (int k = 0; k < 16; ++k) tot += ts[k];
        out[0] = tot / 16.0f;   // / (4 * B * (num_views-1)) = / 16
    }
}

extern "C" void kernel_launch(void* const* d_in, const int* in_sizes, int n_in,
                              void* d_out, int out_size, void* d_ws, size_t ws_size,
                              hipStream_t stream) {
    (void)in_sizes; (void)n_in; (void)out_size; (void)ws_size;
    const float* poses = (const float*)d_in[1];
    const float* masks = (const float*)d_in[2];
    const float* intr  = (const float*)d_in[3];
    const float* depth = (const float*)d_in[4];
    float* ws  = (float*)d_ws;
    float* out = (float*)d_out;

    hipLaunchKernelGGL(prep_kernel,       dim3(1),   dim3(64),  0, stream, poses, intr, ws);
    hipLaunchKernelGGL(mask_count_kernel, dim3(6),   dim3(256), 0, stream, masks, ws);
    hipLaunchKernelGGL(points_kernel,     dim3(224), dim3(256), 0, stream, depth, ws);
    hipLaunchKernelGGL(chamfer_kernel,    dim3(512), dim3(256), 0, stream, masks, ws);
    hipLaunchKernelGGL(finalize_kernel,   dim3(1),   dim3(32),  0, stream, ws, out);
}